// APPNP_Encoder_81690277970589
// MI455X (gfx1250) — compile-verified
//
#include <hip/hip_runtime.h>
#include <stdint.h>

// ---------------------------------------------------------------------------
// APPNP encoder forward for MI455X (gfx1250, wave32).
//   e1 = ego ; e2 = 0.9*A@e1 + ego ; e3 = 0.9*A@e2 + ego   (first SpMM of the
//   reference is A@0 and is skipped). SpMM is L2-resident gather (25.6MB
//   embedding table << 192MB L2); the (col,val) index stream is double-
//   buffered into LDS via gfx1250 async-to-LDS loads (ASYNCcnt). Gathers are
//   issued two-nnz-per-instruction: half-wave x float4 = global_load_b128.
// ---------------------------------------------------------------------------

#define USER_NUM 60000
#define ITEM_NUM 40000
#define N_NODES  (USER_NUM + ITEM_NUM)   // 100000
#define EMB      64
#define NNZ      3200000
#define OMALPHA  0.9f                    // (1 - ALPHA)

#define THREADS   256
#define WPB       8                      // waves per block (wave32)
#define TILE      32                     // (col,val) pairs per staging tile
#define SCAN_T    1024
#define NB        ((N_NODES + SCAN_T - 1) / SCAN_T)   // 98 scan blocks

// ---------------- workspace layout (bytes, 256-aligned) --------------------
#define ALIGN256(x) ((((size_t)(x)) + 255) & ~(size_t)255)
static const size_t WS_ETMP  = 0;                                              // N*EMB f32
static const size_t WS_RPTR  = ALIGN256(WS_ETMP + (size_t)N_NODES * EMB * 4);  // (N+1) i32
static const size_t WS_RCUR  = ALIGN256(WS_RPTR + (size_t)(N_NODES + 1) * 4);  // N i32
static const size_t WS_PAIR  = ALIGN256(WS_RCUR + (size_t)N_NODES * 4);        // NNZ u64
static const size_t WS_BLK   = ALIGN256(WS_PAIR + (size_t)NNZ * 8);            // NB i32
// total ~52 MB

// ---------------------------- CSR build ------------------------------------
__global__ void k_zero_counts(int* __restrict__ cnt) {
  int i = blockIdx.x * blockDim.x + threadIdx.x;
  if (i < N_NODES) cnt[i] = 0;
}

__global__ void k_hist(const int* __restrict__ rows, int* __restrict__ cnt) {
  int i = blockIdx.x * blockDim.x + threadIdx.x;
  if (i < NNZ) atomicAdd(&cnt[rows[i]], 1);
}

__global__ void k_scan_partials(const int* __restrict__ cnt, int* __restrict__ blk) {
  __shared__ int s[SCAN_T];
  int g = blockIdx.x * SCAN_T + threadIdx.x;
  s[threadIdx.x] = (g < N_NODES) ? cnt[g] : 0;
  __syncthreads();
  for (int off = SCAN_T / 2; off > 0; off >>= 1) {
    if (threadIdx.x < off) s[threadIdx.x] += s[threadIdx.x + off];
    __syncthreads();
  }
  if (threadIdx.x == 0) blk[blockIdx.x] = s[0];
}

__global__ void k_scan_blk(int* __restrict__ blk) {   // exclusive scan, NB<=1024
  __shared__ int s[SCAN_T];
  int v = (threadIdx.x < NB) ? blk[threadIdx.x] : 0;
  s[threadIdx.x] = v;
  __syncthreads();
  for (int off = 1; off < SCAN_T; off <<= 1) {
    int t = (threadIdx.x >= off) ? s[threadIdx.x - off] : 0;
    __syncthreads();
    s[threadIdx.x] += t;
    __syncthreads();
  }
  if (threadIdx.x < NB) blk[threadIdx.x] = s[threadIdx.x] - v;
}

__global__ void k_scan_final(const int* __restrict__ cnt, const int* __restrict__ blk,
                             int* __restrict__ row_ptr, int* __restrict__ row_cur) {
  __shared__ int s[SCAN_T];
  int g = blockIdx.x * SCAN_T + threadIdx.x;
  int v = (g < N_NODES) ? cnt[g] : 0;
  s[threadIdx.x] = v;
  __syncthreads();
  for (int off = 1; off < SCAN_T; off <<= 1) {
    int t = (threadIdx.x >= off) ? s[threadIdx.x - off] : 0;
    __syncthreads();
    s[threadIdx.x] += t;
    __syncthreads();
  }
  int excl = s[threadIdx.x] - v + blk[blockIdx.x];
  if (g < N_NODES) { row_ptr[g] = excl; row_cur[g] = excl; }
  if (g == 0) row_ptr[N_NODES] = NNZ;
}

__global__ void k_scatter(const int* __restrict__ rows, const int* __restrict__ cols,
                          const float* __restrict__ vals, int* __restrict__ row_cur,
                          unsigned long long* __restrict__ spair) {
  int i = blockIdx.x * blockDim.x + threadIdx.x;
  if (i < NNZ) {
    int p = atomicAdd(&row_cur[rows[i]], 1);
    unsigned long long pk = ((unsigned long long)__float_as_uint(vals[i]) << 32) |
                            (unsigned long long)(unsigned)cols[i];
    spair[p] = pk;
  }
}

// ---------------------------- e1 = ego -------------------------------------
__global__ void k_copy_ego(const float4* __restrict__ u4, const float4* __restrict__ i4,
                           float4* __restrict__ out4) {
  const int total4 = N_NODES * EMB / 4;   // 1,600,000
  const int user4  = USER_NUM * EMB / 4;  //   960,000 (float4-aligned boundary)
  int i = blockIdx.x * blockDim.x + threadIdx.x;
  if (i < total4) out4[i] = (i < user4) ? u4[i] : i4[i - user4];
}

// ------------------------------- SpMM --------------------------------------
// One wave per row. Two nnz are processed per gather instruction: lanes 0-15
// gather row a, lanes 16-31 gather row b, each lane a float4 (b128). Each
// half-wave keeps a float4 partial for output columns {4*hl..4*hl+3}; the two
// halves are merged with __shfl_xor(16) at the end. (col,val) tiles stream
// into a per-wave LDS double buffer via gfx1250 global_load_async_to_lds_b64,
// pipelined on ASYNCcnt.
__device__ __forceinline__ void async_pair(uint32_t lds_addr,
                                           const unsigned long long* base,
                                           uint32_t byte_off) {
  asm volatile("global_load_async_to_lds_b64 %0, %1, %2"
               :: "v"(lds_addr), "v"(byte_off), "s"(base) : "memory");
}

__global__ __launch_bounds__(THREADS)
void k_spmm(const int* __restrict__ row_ptr,
            const unsigned long long* __restrict__ spair,
            const float* __restrict__ e_in,
            const float* __restrict__ uemb, const float* __restrict__ iemb,
            float* __restrict__ e_out) {
  __shared__ unsigned long long stage[WPB][2][TILE];

  const int lane = threadIdx.x & 31;
  const int wave = threadIdx.x >> 5;
  const int half = lane >> 4;              // 0: even nnz, 1: odd nnz
  const int hl   = lane & 15;              // column group: cols 4*hl..4*hl+3
  const int row  = blockIdx.x * WPB + wave;
  if (row >= N_NODES) return;

  const int beg = row_ptr[row];
  const int end = row_ptr[row + 1];
  const int nnz = end - beg;
  const int nt  = (nnz + TILE - 1) / TILE;

  // LDS byte addresses of this lane's two staging slots (generic->LDS offset
  // by 32-bit truncation: shared aperture keeps DS offset in addr[31:0]).
  const uint32_t lds0 = (uint32_t)(uintptr_t)&stage[wave][0][lane];
  const uint32_t lds1 = (uint32_t)(uintptr_t)&stage[wave][1][lane];

  float4 acc = make_float4(0.f, 0.f, 0.f, 0.f);

  if (nt > 0) {                                   // prime tile 0
    if (lane < min(TILE, nnz))
      async_pair(lds0, spair, (uint32_t)((beg + lane) * 8u));
  }

  for (int t = 0; t < nt; ++t) {
    const int base = beg + t * TILE;
    const int cnt  = min(TILE, end - base);
    if (t + 1 < nt) {                             // issue tile t+1, then wait t
      const int nb = base + TILE;
      const int nc = min(TILE, end - nb);
      const uint32_t dst = ((t + 1) & 1) ? lds1 : lds0;
      if (lane < nc)
        async_pair(dst, spair, (uint32_t)((nb + lane) * 8u));
      asm volatile("s_wait_asynccnt 0x1" ::: "memory");
    } else {
      asm volatile("s_wait_asynccnt 0x0" ::: "memory");
    }

    const unsigned long long* buf = stage[wave][t & 1];
    const int pairs = (cnt + 1) >> 1;
    for (int k = 0; k < pairs; ++k) {
      const int idx = 2 * k + half;               // this half-wave's nnz
      const unsigned long long p = buf[(idx < cnt) ? idx : (cnt - 1)];
      const int   col = (int)(uint32_t)p;
      const float v   = (idx < cnt) ? __uint_as_float((uint32_t)(p >> 32)) : 0.f;
      // one b128 per lane: 32 lanes cover two full 256B embedding rows;
      // e_in table is L2-resident (25.6MB << 192MB L2)
      const float4 g = *(const float4*)(e_in + (size_t)col * EMB + hl * 4);
      acc.x += v * g.x;
      acc.y += v * g.y;
      acc.z += v * g.z;
      acc.w += v * g.w;
    }
  }

  // merge the two half-wave partials (lane l <- lane l^16)
  acc.x += __shfl_xor(acc.x, 16, 32);
  acc.y += __shfl_xor(acc.y, 16, 32);
  acc.z += __shfl_xor(acc.z, 16, 32);
  acc.w += __shfl_xor(acc.w, 16, 32);

  // fused combine: e_out = 0.9*msg + ego  (ego = concat(user, item) row)
  if (half == 0) {
    const size_t oidx = (size_t)row * EMB + hl * 4;
    const float* ego = (row < USER_NUM)
                         ? (uemb + oidx)
                         : (iemb + (oidx - (size_t)USER_NUM * EMB));
    const float4 eg = *(const float4*)ego;
    float4 r;
    r.x = OMALPHA * acc.x + eg.x;
    r.y = OMALPHA * acc.y + eg.y;
    r.z = OMALPHA * acc.z + eg.z;
    r.w = OMALPHA * acc.w + eg.w;
    *(float4*)(e_out + oidx) = r;
  }
}

// ------------------------------ launcher -----------------------------------
extern "C" void kernel_launch(void* const* d_in, const int* in_sizes, int n_in,
                              void* d_out, int out_size, void* d_ws, size_t ws_size,
                              hipStream_t stream) {
  const int*   rows = (const int*)  d_in[0];
  const int*   cols = (const int*)  d_in[1];
  const float* vals = (const float*)d_in[2];
  const float* uemb = (const float*)d_in[3];
  const float* iemb = (const float*)d_in[4];
  float* out = (float*)d_out;

  char* ws = (char*)d_ws;
  float*              e_tmp   = (float*)             (ws + WS_ETMP);
  int*                row_ptr = (int*)               (ws + WS_RPTR);
  int*                row_cur = (int*)               (ws + WS_RCUR);
  unsigned long long* spair   = (unsigned long long*)(ws + WS_PAIR);
  int*                blk     = (int*)               (ws + WS_BLK);

  const int gN    = (N_NODES + THREADS - 1) / THREADS;
  const int gNNZ  = (NNZ + THREADS - 1) / THREADS;
  const int gCopy = (N_NODES * EMB / 4 + THREADS - 1) / THREADS;
  const int gRows = (N_NODES + WPB - 1) / WPB;      // 12500 blocks, 1 wave/row

  // --- build CSR (counting sort by row; structure reused for both SpMMs) ---
  k_zero_counts  <<<gN,   THREADS, 0, stream>>>(row_cur);   // row_cur as counts
  k_hist         <<<gNNZ, THREADS, 0, stream>>>(rows, row_cur);
  k_scan_partials<<<NB,   SCAN_T,  0, stream>>>(row_cur, blk);
  k_scan_blk     <<<1,    SCAN_T,  0, stream>>>(blk);
  k_scan_final   <<<NB,   SCAN_T,  0, stream>>>(row_cur, blk, row_ptr, row_cur);
  k_scatter      <<<gNNZ, THREADS, 0, stream>>>(rows, cols, vals, row_cur, spair);

  // --- e1 = ego -> d_out ---
  k_copy_ego<<<gCopy, THREADS, 0, stream>>>((const float4*)uemb, (const float4*)iemb,
                                            (float4*)out);

  // --- e2 = 0.9*A@e1 + ego -> ws ; e3 = 0.9*A@e2 + ego -> d_out ---
  k_spmm<<<gRows, THREADS, 0, stream>>>(row_ptr, spair, out,   uemb, iemb, e_tmp);
  k_spmm<<<gRows, THREADS, 0, stream>>>(row_ptr, spair, e_tmp, uemb, iemb, out);
}